// OctreeGroupNorm_84507776516600
// MI455X (gfx1250) — compile-verified
//
#include <hip/hip_runtime.h>

typedef float v2f __attribute__((ext_vector_type(2)));
typedef float v8f __attribute__((ext_vector_type(8)));

#define C_CH   64
#define GROUPS 16
#define CPG    4
#define EPS    1e-5f

// workspace layout in floats
#define WS_CNT 0                 // [16]   per-batch row counts
#define WS_S1  16                // [16*64] per-batch per-channel sums
#define WS_S2  (16 + 1024)       // [16*64] per-batch per-channel sum of squares
#define WS_M   (16 + 2048)       // [16*16] per-(batch,group) mean
#define WS_IS  (16 + 2048 + 256) // [16*16] per-(batch,group) inv-std
#define WS_ZERO_COUNT (16 + 2048)

__global__ void ogn_zero(float* __restrict__ ws) {
  int i = blockIdx.x * blockDim.x + threadIdx.x;
  const int stride = gridDim.x * blockDim.x;
  for (; i < WS_ZERO_COUNT; i += stride) ws[i] = 0.0f;
}

// Pass 1: per-(batch, channel) sum and sum-of-squares via V_WMMA_F32_16X16X4_F32.
// A = ones(16x4)  =>  D[m][n] = sum_k B[k][n] + C[m][n]  (column sums, uniform over m).
__global__ __launch_bounds__(256) void ogn_stats(const float* __restrict__ data,
                                                 const int* __restrict__ bid,
                                                 float* __restrict__ ws, int N) {
  float* CNTp = ws + WS_CNT;
  float* S1   = ws + WS_S1;
  float* S2   = ws + WS_S2;

  const int lane = threadIdx.x & 31;
  const int half = lane >> 4;   // which K pair this lane's loads feed
  const int col  = lane & 15;   // matrix column (channel within 16-wide tile)
  const int waveId = blockIdx.x * (blockDim.x >> 5) + (threadIdx.x >> 5);
  const int nWaves = gridDim.x * (blockDim.x >> 5);

  const int nChunks = N >> 2;                       // 4 rows per chunk
  const int per = (nChunks + nWaves - 1) / nWaves;  // contiguous range per wave
  const int begin = waveId * per;
  int end = begin + per;
  if (end > nChunks) end = nChunks;

  const v2f onesA = {1.0f, 1.0f};
  const v8f zero8 = {};
  v8f accS[4] = {};  // channel sums, 4 tiles of 16 channels
  v8f accQ[4] = {};  // channel sums of squares
  int waveRows = 0;
  int curB = 0;

  if (begin < end) {
    curB = __builtin_amdgcn_readfirstlane(bid[begin << 2]);

    for (int cs = begin; cs < end; ++cs) {
      const int row0 = cs << 2;
      const int ra = row0 + half;       // rows for K = half
      const int rb = ra + 2;            // rows for K = half + 2
      const int ba = bid[ra];
      const int bb = bid[rb];
      const int bFirst = __builtin_amdgcn_readfirstlane(ba);
      const int bLast  = __builtin_amdgcn_readlane(bb, 31);

      float va[4], vb[4];
#pragma unroll
      for (int g = 0; g < 4; ++g) {
        va[g] = data[(size_t)ra * C_CH + g * 16 + col];
        vb[g] = data[(size_t)rb * C_CH + g * 16 + col];
      }

      for (int b = bFirst;; ++b) {
        if (b != curB) {
          // flush accumulators for previous batch
          if (half == 0) {
#pragma unroll
            for (int g = 0; g < 4; ++g) {
              atomicAdd(&S1[curB * C_CH + g * 16 + col], accS[g][0]);
              atomicAdd(&S2[curB * C_CH + g * 16 + col], accQ[g][0]);
            }
          }
          if (lane == 0 && waveRows) atomicAdd(&CNTp[curB], (float)waveRows);
          waveRows = 0;
#pragma unroll
          for (int g = 0; g < 4; ++g) { accS[g] = zero8; accQ[g] = zero8; }
          curB = b;
        }

        const bool ma = (ba == b);
        const bool mb = (bb == b);
        const unsigned mka = __builtin_amdgcn_ballot_w32(ma);
        const unsigned mkb = __builtin_amdgcn_ballot_w32(mb);
        waveRows += (int)((__builtin_popcount(mka) + __builtin_popcount(mkb)) >> 4);

#pragma unroll
        for (int g = 0; g < 4; ++g) {
          v2f Bv;
          Bv.x = ma ? va[g] : 0.0f;
          Bv.y = mb ? vb[g] : 0.0f;
          v2f Bq;
          Bq.x = Bv.x * Bv.x;
          Bq.y = Bv.y * Bv.y;
          accS[g] = __builtin_amdgcn_wmma_f32_16x16x4_f32(
              false, onesA, false, Bv, (short)0, accS[g], false, false);
          accQ[g] = __builtin_amdgcn_wmma_f32_16x16x4_f32(
              false, onesA, false, Bq, (short)0, accQ[g], false, false);
        }
        if (b >= bLast) break;
      }
    }

    // final flush
    if (half == 0) {
#pragma unroll
      for (int g = 0; g < 4; ++g) {
        atomicAdd(&S1[curB * C_CH + g * 16 + col], accS[g][0]);
        atomicAdd(&S2[curB * C_CH + g * 16 + col], accQ[g][0]);
      }
    }
    if (lane == 0 && waveRows) atomicAdd(&CNTp[curB], (float)waveRows);
  }

  // tail rows (N % 4) handled by wave 0 with plain atomics
  if (waveId == 0) {
    for (int r = nChunks << 2; r < N; ++r) {
      const int b = bid[r];
      const float x0 = data[(size_t)r * C_CH + lane];
      const float x1 = data[(size_t)r * C_CH + lane + 32];
      atomicAdd(&S1[b * C_CH + lane], x0);
      atomicAdd(&S2[b * C_CH + lane], x0 * x0);
      atomicAdd(&S1[b * C_CH + lane + 32], x1);
      atomicAdd(&S2[b * C_CH + lane + 32], x1 * x1);
      if (lane == 0) atomicAdd(&CNTp[b], 1.0f);
    }
  }
}

// Pass 2: reduce CPG channels per group, compute mean & inv-std (256 = 16 batches x 16 groups).
__global__ void ogn_finalize(float* __restrict__ ws) {
  const int t = threadIdx.x;      // 0..255
  const int b = t >> 4;
  const int g = t & 15;
  const float* S1 = ws + WS_S1;
  const float* S2 = ws + WS_S2;

  float s1 = 0.0f, s2 = 0.0f;
#pragma unroll
  for (int j = 0; j < CPG; ++j) {
    s1 += S1[b * C_CH + g * CPG + j];
    s2 += S2[b * C_CH + g * CPG + j];
  }
  const float count = ws[WS_CNT + b] * (float)CPG;   // n_b * CPG
  const float invc  = 1.0f / (count + EPS);
  const float m     = s1 * invc;                     // matches reference group-adjusted mean
  const float var   = (s2 - 2.0f * m * s1 + count * m * m) * invc;
  const float istd  = rsqrtf(var + EPS);
  ws[WS_M + t]  = m;
  ws[WS_IS + t] = istd;
}

// Pass 3: normalize + affine. Each thread owns 4 consecutive channels == exactly one group,
// so float4 (b128) loads/stores and a single (m, inv_std) pair per row.
__global__ __launch_bounds__(256) void ogn_apply(const float4* __restrict__ data4,
                                                 const int* __restrict__ bid,
                                                 const float4* __restrict__ w4,
                                                 const float4* __restrict__ bias4,
                                                 const float* __restrict__ ws,
                                                 float4* __restrict__ out4, int n4) {
  const float* M  = ws + WS_M;
  const float* IS = ws + WS_IS;
  const int t0 = blockIdx.x * blockDim.x + threadIdx.x;
  const int stride = gridDim.x * blockDim.x;   // multiple of 16 -> group index fixed per thread
  const int t = t0 & 15;                       // group id / float4 slot within row
  const float4 wv = w4[t];
  const float4 bv = bias4[t];

  for (int i = t0; i < n4; i += stride) {
    const int row = i >> 4;
    const int b = bid[row];
    const float m = M[b * GROUPS + t];
    const float s = IS[b * GROUPS + t];
    const float4 x = data4[i];
    float4 o;
    o.x = (x.x - m) * s * wv.x + bv.x;
    o.y = (x.y - m) * s * wv.y + bv.y;
    o.z = (x.z - m) * s * wv.z + bv.z;
    o.w = (x.w - m) * s * wv.w + bv.w;
    out4[i] = o;
  }
}

extern "C" void kernel_launch(void* const* d_in, const int* in_sizes, int n_in,
                              void* d_out, int out_size, void* d_ws, size_t ws_size,
                              hipStream_t stream) {
  const float* data = (const float*)d_in[0];
  const int*   bid  = (const int*)d_in[1];
  const float* w    = (const float*)d_in[2];
  const float* bias = (const float*)d_in[3];
  float* out = (float*)d_out;
  float* ws  = (float*)d_ws;

  const int N = in_sizes[0] / C_CH;

  ogn_zero<<<8, 256, 0, stream>>>(ws);
  ogn_stats<<<512, 256, 0, stream>>>(data, bid, ws, N);
  ogn_finalize<<<1, 256, 0, stream>>>(ws);

  const int n4 = N * (C_CH / 4);
  ogn_apply<<<1024, 256, 0, stream>>>((const float4*)data, bid,
                                      (const float4*)w, (const float4*)bias,
                                      ws, (float4*)out, n4);
}